// mGRU_8340826488966
// MI455X (gfx1250) — compile-verified
//
#include <hip/hip_runtime.h>
#include <math.h>

#define NHID 1024
#define NVOC 31
#define NBATCH 64
#define NSEQ 512
#define G3 3072                         // 3*NHID
#define NROWS (NBATCH*NSEQ)

#define SB_FLOATS ((NHID/2) * 48 * 2)   // LDS-resident Whh slice: 49152 floats (192 KB)
#define RED_FLOATS (32 * 128)           // K-split reduction scratch: 4096 floats (16 KB)
#define SMEM_BYTES ((SB_FLOATS + RED_FLOATS) * 4)   // 212992 B < 320 KB/WGP

typedef float v2f __attribute__((ext_vector_type(2)));
typedef float v8f __attribute__((ext_vector_type(8)));
typedef unsigned int u32x4 __attribute__((ext_vector_type(4)));
typedef int i32x8 __attribute__((ext_vector_type(8)));
typedef int i32x4 __attribute__((ext_vector_type(4)));

__device__ __forceinline__ v8f wmma4(v2f a, v2f b, v8f c) {
  // V_WMMA_F32_16X16X4_F32 : D = A(16x4) * B(4x16) + C(16x16), full fp32
  return __builtin_amdgcn_wmma_f32_16x16x4_f32(false, a, false, b, (short)0, c,
                                               false, false);
}

// ---------------------------------------------------------------------------
// Pair-packed K-major weights: P[(k/2)*G3 + gj][2] = { W[gj][k], W[gj][k+1] }.
// A lane's WMMA B-fragment becomes ONE aligned 8B load; a half-wave covers a
// fully contiguous 128B region (perfect coalescing / LDS bank usage).
// ---------------------------------------------------------------------------
__global__ void pack_pairs_kernel(const float* __restrict__ W,
                                  float* __restrict__ P) {
  int idx = blockIdx.x * 256 + threadIdx.x;        // over (NHID/2)*G3
  if (idx >= (NHID / 2) * G3) return;
  int p  = idx / G3;
  int gj = idx - p * G3;
  v2f v = {W[(size_t)gj * NHID + 2 * p], W[(size_t)gj * NHID + 2 * p + 1]};
  *(v2f*)(P + (size_t)idx * 2) = v;                // write-coalesced b64
}

// ---------------------------------------------------------------------------
// embW[tok][gj] = bih0[gj] + sum_k emb[tok][k]*Wih0[gj][k]
// Tokens take only 31 values -> layer-0 input gates collapse to this table.
// ---------------------------------------------------------------------------
__global__ void embw_kernel(const float* __restrict__ emb,
                            const float* __restrict__ Wih0,
                            const float* __restrict__ bih0,
                            float* __restrict__ embW) {
  int idx = blockIdx.x * 256 + threadIdx.x;        // over 31*3072
  if (idx >= NVOC * G3) return;
  int tok = idx / G3;
  int gj  = idx - tok * G3;
  const float* e = emb + (size_t)tok * NHID;
  const float* w = Wih0 + (size_t)gj * NHID;
  float s = bih0[gj];
  for (int k = 0; k < NHID; k += 4)
    s += e[k]*w[k] + e[k+1]*w[k+1] + e[k+2]*w[k+2] + e[k+3]*w[k+3];
  embW[idx] = s;
}

// ---------------------------------------------------------------------------
// Counter-based grid barrier (all 64 blocks are co-resident by construction:
// one 256-thread block per WGP with ~208KB LDS).
// ---------------------------------------------------------------------------
__device__ __forceinline__ void grid_sync(unsigned* ctr, unsigned target) {
  __threadfence();
  __syncthreads();
  if (threadIdx.x == 0) {
    __hip_atomic_fetch_add(ctr, 1u, __ATOMIC_RELEASE, __HIP_MEMORY_SCOPE_AGENT);
    while (__hip_atomic_load(ctr, __ATOMIC_ACQUIRE, __HIP_MEMORY_SCOPE_AGENT) <
           target)
      __builtin_amdgcn_s_sleep(2);
  }
  __syncthreads();
}

// ---------------------------------------------------------------------------
// Stage this block's pair-packed Whh slice (3D tile: X=32 contiguous floats,
// Y=3 gates stride 2048 elem, Z=512 pairs stride 6144 elem) into LDS at
// offset 0 via the Tensor Data Mover. TDM's natural fill order (X,Y,Z ->
// contiguous LDS) reproduces exactly the sB[(p*48+g*16+j)*2] layout the WMMA
// loop consumes.
// ---------------------------------------------------------------------------
__device__ __forceinline__ void stage_whh_lds(const float* WhhP, int jbase,
                                              float* sB, int tid, int wave) {
#if __has_builtin(__builtin_amdgcn_tensor_load_to_lds) && \
    __has_builtin(__builtin_amdgcn_s_wait_tensorcnt)
  if (wave == 0) {                                  // one TDM issue per block
    const unsigned long long ga =
        (unsigned long long)(const void*)(WhhP + (size_t)jbase * 2);
    u32x4 g0;
    g0[0] = 0x1u;                                   // count=1, user descriptor
    g0[1] = 0u;                                     // lds_addr: dynamic LDS base
    g0[2] = (unsigned)(ga & 0xFFFFFFFFu);           // global_addr[31:0]
    g0[3] = (unsigned)((ga >> 32) & 0x1FFFFFFu)     // global_addr[56:32]
            | (2u << 30);                           // type = 2 ("image")
    i32x8 g1;
    g1[0] = (2 << 16);                              // data_size = 4B
    g1[1] = (int)(2048u << 16);                     // tensor_dim0 lo16 = 2048
    g1[2] = (3 << 16);                              // dim0 hi=0 | tensor_dim1 lo16=3
    g1[3] = (32 << 16);                             // dim1 hi=0 | tile_dim0=32
    g1[4] = 3 | (512 << 16);                        // tile_dim1=3, tile_dim2=512
    g1[5] = 2048;                                   // tensor_dim0_stride lo32
    g1[6] = (int)(6144u << 16);                     // t0s hi=0 | tensor_dim1_stride lo16
    g1[7] = 0;                                      // tensor_dim1_stride hi
    i32x4 g2 = {512, 1, 6144, 0};                   // tensor_dim2/3, dim2_stride, tile_dim3=0
    i32x4 g3 = {0, 0, 0, 0};
    i32x8 g4 = {0, 0, 0, 0, 0, 0, 0, 0};            // unused trailing group
    __builtin_amdgcn_tensor_load_to_lds(g0, g1, g2, g3, g4, 0);
    __builtin_amdgcn_s_wait_tensorcnt(0);           // TENSORcnt is per-wave
  }
#else
  for (int f = tid; f < (NHID / 2) * 3 * 8; f += 256) {   // 12288 float4's
    int pg = f >> 3, q = f & 7;
    int p = pg / 3, g = pg - 3 * p;
    float4 v = *(const float4*)(WhhP +
        ((size_t)p * G3 + g * NHID + jbase) * 2 + q * 4);
    *(float4*)(sB + (p * 48 + g * 16) * 2 + q * 4) = v;
  }
#endif
  __syncthreads();                                  // consumers in all waves
}

// ---------------------------------------------------------------------------
// Persistent GRU layer: one kernel runs all 512 timesteps.
//   grid = 64 blocks (16 hidden cols each), block = 256 threads = 8 waves:
//   wave = (K-half 0/1) x (batch tile 0..3). Whh slice lives in LDS for the
//   whole kernel (TDM-staged); K-split partials reduce through LDS each step;
//   blocks synchronize between steps with the counter barrier (double-
//   buffered h). Layer 1 fuses y0_t @ Wih1^T into the same K loop (r/z
//   combined with the h-side; n-gate keeps xn/hn separate: tanh(xn + r*hn)).
// ---------------------------------------------------------------------------
template <bool L1>
__global__ __launch_bounds__(256)
void gru_layer_kernel(float* __restrict__ hA, float* __restrict__ hB,
                      const float* __restrict__ WhhP,  // pair-packed [512][3072][2]
                      const float* __restrict__ bhh,
                      const float* __restrict__ WihP,  // pair-packed (L1 only)
                      const float* __restrict__ bih,   // (L1 only)
                      const float* __restrict__ xsrc,  // L0: embW, L1: y0
                      const int*   __restrict__ tok,   // L0: token ids [B,S]
                      float* __restrict__ y,           // [B,S,NHID]
                      unsigned* __restrict__ ctr) {
  extern __shared__ float smem[];
  float* sB  = smem;                 // [pair p][gate][j][2] : 49152 floats
  float* red = smem + SB_FLOATS;     // [comp 0..31][128]    : 4096 floats

  const int tid   = threadIdx.x;
  const int lane  = tid & 31;
  const int wave  = tid >> 5;        // 0..7
  const int mt    = wave & 3;        // batch tile
  const int ks    = wave >> 2;       // K-split half
  const int jbase = blockIdx.x * 16;
  const int n     = lane & 15;
  const int kh    = lane >> 4;       // K-pair selector inside a 4-chunk

  stage_whh_lds(WhhP, jbase, sB, tid, wave);

  const int kbeg = ks * (NHID / 2);
  const int kend = kbeg + NHID / 2;
  const int arow = mt * 16 + n;      // A-operand row (m = lane&15)
  const int j    = jbase + n;

  for (int t = 0; t < NSEQ; ++t) {
    const float* h_in  = (t & 1) ? hB : hA;
    float*       h_out = (t & 1) ? hA : hB;

    v8f ar = {}, az = {}, ahn = {}, axn = {};
    const float* aptr = h_in + (size_t)arow * NHID + 2 * kh;
    const float* xptr = L1 ? xsrc + ((size_t)arow * NSEQ + t) * NHID + 2 * kh
                           : nullptr;

#pragma unroll 4
    for (int kk = kbeg; kk < kend; kk += 4) {
      v2f a = *(const v2f*)(aptr + kk);
      const int p2 = (kk + 2 * kh) >> 1;               // pair index (even k)
      const float* sb = sB + ((size_t)p2 * 48 + n) * 2;
      v2f b0 = *(const v2f*)(sb);
      v2f b1 = *(const v2f*)(sb + 32);
      v2f b2 = *(const v2f*)(sb + 64);
      ar  = wmma4(a, b0, ar);
      az  = wmma4(a, b1, az);
      ahn = wmma4(a, b2, ahn);
      if (L1) {
        v2f ax = *(const v2f*)(xptr + kk);
        const float* gp = WihP + ((size_t)p2 * G3 + jbase + n) * 2;
        v2f c0 = *(const v2f*)(gp);
        v2f c1 = *(const v2f*)(gp + 2 * NHID);
        v2f c2 = *(const v2f*)(gp + 4 * NHID);
        ar  = wmma4(ax, c0, ar);
        az  = wmma4(ax, c1, az);
        axn = wmma4(ax, c2, axn);
      }
    }

    // ---- K-split reduction through LDS (component-major: conflict-free) ----
    const int slot = mt * 32 + lane;                    // 0..127
    if (ks == 1) {
#pragma unroll
      for (int c = 0; c < 8; ++c) {
        red[(c     ) * 128 + slot] = ar[c];
        red[(c +  8) * 128 + slot] = az[c];
        red[(c + 16) * 128 + slot] = ahn[c];
        if (L1) red[(c + 24) * 128 + slot] = axn[c];
      }
    }
    __syncthreads();

    if (ks == 0) {
#pragma unroll
      for (int c = 0; c < 8; ++c) {
        ar[c]  += red[(c     ) * 128 + slot];
        az[c]  += red[(c +  8) * 128 + slot];
        ahn[c] += red[(c + 16) * 128 + slot];
        if (L1) axn[c] += red[(c + 24) * 128 + slot];
      }
      const float bh_r = bhh[j], bh_z = bhh[NHID + j], bh_n = bhh[2*NHID + j];
      float bi_r = 0.f, bi_z = 0.f, bi_n = 0.f;
      if (L1) { bi_r = bih[j]; bi_z = bih[NHID + j]; bi_n = bih[2*NHID + j]; }

#pragma unroll
      for (int r = 0; r < 8; ++r) {
        const int m = mt * 16 + r + 8 * kh;             // batch row
        float xr, xz, xn;
        if (L1) {
          xr = bi_r; xz = bi_z; xn = axn[r] + bi_n;     // x-GEMM fused in accs
        } else {
          const int tv = tok[(size_t)m * NSEQ + t];
          const float* eb = xsrc + (size_t)tv * G3;     // embW (has bih0)
          xr = eb[j]; xz = eb[NHID + j]; xn = eb[2*NHID + j];
        }
        const float rpre = ar[r] + bh_r + xr;
        const float zpre = az[r] + bh_z + xz;
        const float hnp  = ahn[r] + bh_n;
        const float rg = 1.f / (1.f + __expf(-rpre));
        const float zg = 1.f / (1.f + __expf(-zpre));
        const float ng = tanhf(xn + rg * hnp);
        const float hold = h_in[(size_t)m * NHID + j];
        const float hnew = (1.f - zg) * ng + zg * hold;
        h_out[(size_t)m * NHID + j] = hnew;
        y[((size_t)m * NSEQ + t) * NHID + j] = hnew;
      }
    }

    if (t + 1 < NSEQ)
      grid_sync(ctr, (unsigned)(gridDim.x * (t + 1)));  // release h_out
  }
}

// ---------------------------------------------------------------------------
// BatchNorm statistics over flattened (B*S, NH).
// ---------------------------------------------------------------------------
__global__ void bn_partial_kernel(const float* __restrict__ y1,
                                  float* __restrict__ psum,
                                  float* __restrict__ psq) {
  const int c     = blockIdx.x * 256 + threadIdx.x;   // channel, gridDim.x=4
  const int slice = blockIdx.y;                       // 64 row slices
  const int rows  = NROWS / 64;
  float s = 0.f, s2 = 0.f;
  const float* p = y1 + (size_t)slice * rows * NHID + c;
  for (int r = 0; r < rows; ++r) {
    float v = p[(size_t)r * NHID];
    s += v; s2 += v * v;
  }
  psum[(size_t)slice * NHID + c] = s;
  psq [(size_t)slice * NHID + c] = s2;
}

__global__ void bn_final_kernel(const float* __restrict__ psum,
                                const float* __restrict__ psq,
                                const float* __restrict__ gamma,
                                float* __restrict__ mu,
                                float* __restrict__ rsg) {
  const int c = blockIdx.x * 256 + threadIdx.x;
  float s = 0.f, s2 = 0.f;
  for (int k = 0; k < 64; ++k) {
    s  += psum[(size_t)k * NHID + c];
    s2 += psq [(size_t)k * NHID + c];
  }
  const float m   = s / (float)NROWS;
  const float var = s2 / (float)NROWS - m * m;
  mu[c]  = m;
  rsg[c] = rsqrtf(var + 1e-5f) * gamma[c];
}

// Fold BN into the vocab projection. Pair-packed W': WpP[(c/2)*32+v][2].
__global__ void wout_prep_kernel(const float* __restrict__ Wout,
                                 const float* __restrict__ rsg,
                                 float* __restrict__ WpP) {
  int idx = blockIdx.x * 256 + threadIdx.x;           // over 512*32
  if (idx >= (NHID / 2) * 32) return;
  int p = idx >> 5, v = idx & 31;
  int c0 = 2 * p;
  v2f w = {0.f, 0.f};
  if (v < NVOC) {
    w.x = rsg[c0]     * Wout[(size_t)v * NHID + c0];
    w.y = rsg[c0 + 1] * Wout[(size_t)v * NHID + c0 + 1];
  }
  *(v2f*)(WpP + (size_t)idx * 2) = w;
}

__global__ void bout_prep_kernel(const float* __restrict__ Wout,
                                 const float* __restrict__ bout,
                                 const float* __restrict__ beta,
                                 const float* __restrict__ mu,
                                 const float* __restrict__ rsg,
                                 float* __restrict__ bp) {
  const int v = threadIdx.x;                          // 32 threads
  float s = 0.f;
  if (v < NVOC) {
    s = bout[v];
    for (int c = 0; c < NHID; ++c)
      s += (beta[c] - mu[c] * rsg[c]) * Wout[(size_t)v * NHID + c];
  }
  if (v < 32) bp[v] = s;
}

// out[row][v] = sum_c y1[row][c] * W'[c][v] + b'[v] via fp32 WMMA.
__global__ __launch_bounds__(128)
void out_gemm_kernel(const float* __restrict__ y1,
                     const float* __restrict__ WpP,
                     const float* __restrict__ bp,
                     float* __restrict__ out) {
  const int lane = threadIdx.x & 31;
  const int w    = threadIdx.x >> 5;
  const int mw   = w >> 1, nw = w & 1;
  const int mtile = blockIdx.x * 2 + mw;
  const int n  = lane & 15;
  const int kh = lane >> 4;
  const int arow = mtile * 16 + n;
  v8f acc = {};
  const float* ap = y1 + (size_t)arow * NHID + 2 * kh;
#pragma unroll 8
  for (int kk = 0; kk < NHID; kk += 4) {
    v2f a = *(const v2f*)(ap + kk);
    const int p2 = (kk + 2 * kh) >> 1;
    v2f b = *(const v2f*)(WpP + ((size_t)p2 * 32 + nw * 16 + n) * 2);
    acc = wmma4(a, b, acc);
  }
#pragma unroll
  for (int r = 0; r < 8; ++r) {
    const int m = mtile * 16 + r + 8 * kh;
    const int v = nw * 16 + n;
    if (v < NVOC) out[(size_t)m * NVOC + v] = acc[r] + bp[v];
  }
}

// ---------------------------------------------------------------------------
extern "C" void kernel_launch(void* const* d_in, const int* in_sizes, int n_in,
                              void* d_out, int out_size, void* d_ws,
                              size_t ws_size, hipStream_t stream) {
  const int*   x     = (const int*)  d_in[0];
  const float* h0    = (const float*)d_in[1];
  const float* emb   = (const float*)d_in[2];
  const float* Wih0  = (const float*)d_in[3];
  const float* Whh0  = (const float*)d_in[4];
  const float* bih0  = (const float*)d_in[5];
  const float* bhh0  = (const float*)d_in[6];
  const float* Wih1  = (const float*)d_in[7];
  const float* Whh1  = (const float*)d_in[8];
  const float* bih1  = (const float*)d_in[9];
  const float* bhh1  = (const float*)d_in[10];
  const float* gamma = (const float*)d_in[11];
  const float* beta  = (const float*)d_in[12];
  const float* Wout  = (const float*)d_in[13];
  const float* bout  = (const float*)d_in[14];
  float* out = (float*)d_out;

  uint8_t* p = (uint8_t*)d_ws;
  auto alloc = [&](size_t bytes) -> void* {
    void* r = (void*)p;
    p += (bytes + 255) & ~(size_t)255;
    return r;
  };
  float*    y0    = (float*)alloc((size_t)NROWS * NHID * 4);   // 134 MB
  float*    y1    = (float*)alloc((size_t)NROWS * NHID * 4);   // 134 MB
  float*    WhhP0 = (float*)alloc((size_t)NHID * G3 * 4);      // 12.6 MB
  float*    WhhP1 = (float*)alloc((size_t)NHID * G3 * 4);
  float*    WihP1 = (float*)alloc((size_t)NHID * G3 * 4);
  float*    embW  = (float*)alloc((size_t)NVOC * G3 * 4);
  float*    hA    = (float*)alloc((size_t)NBATCH * NHID * 4);
  float*    hB    = (float*)alloc((size_t)NBATCH * NHID * 4);
  float*    psum  = (float*)alloc((size_t)64 * NHID * 4);
  float*    psq   = (float*)alloc((size_t)64 * NHID * 4);
  float*    mu    = (float*)alloc((size_t)NHID * 4);
  float*    rsg   = (float*)alloc((size_t)NHID * 4);
  float*    WpP   = (float*)alloc((size_t)(NHID / 2) * 32 * 2 * 4);
  float*    bp    = (float*)alloc(32 * 4);
  unsigned* ctr   = (unsigned*)alloc(256);

  // --- one-time prep ---
  {
    const int nP = ((NHID / 2) * G3 + 255) / 256;
    pack_pairs_kernel<<<nP, 256, 0, stream>>>(Whh0, WhhP0);
    pack_pairs_kernel<<<nP, 256, 0, stream>>>(Whh1, WhhP1);
    pack_pairs_kernel<<<nP, 256, 0, stream>>>(Wih1, WihP1);
    const int nE = (NVOC * G3 + 255) / 256;
    embw_kernel<<<nE, 256, 0, stream>>>(emb, Wih0, bih0, embW);
  }

  // --- layer 0: persistent kernel over all 512 steps ---
  (void)hipMemcpyAsync(hA, h0, (size_t)NBATCH * NHID * 4,
                       hipMemcpyDeviceToDevice, stream);
  (void)hipMemsetAsync(ctr, 0, sizeof(unsigned), stream);
  gru_layer_kernel<false><<<NHID / 16, 256, SMEM_BYTES, stream>>>(
      hA, hB, WhhP0, bhh0, nullptr, nullptr, embW, x, y0, ctr);

  // --- layer 1: persistent kernel, x-side GEMM fused ---
  (void)hipMemcpyAsync(hA, h0 + (size_t)NBATCH * NHID, (size_t)NBATCH * NHID * 4,
                       hipMemcpyDeviceToDevice, stream);
  (void)hipMemsetAsync(ctr, 0, sizeof(unsigned), stream);
  gru_layer_kernel<true><<<NHID / 16, 256, SMEM_BYTES, stream>>>(
      hA, hB, WhhP1, bhh1, WihP1, bih1, y0, nullptr, y1, ctr);

  // --- BatchNorm stats + folded vocab projection ---
  bn_partial_kernel<<<dim3(NHID / 256, 64), 256, 0, stream>>>(y1, psum, psq);
  bn_final_kernel<<<NHID / 256, 256, 0, stream>>>(psum, psq, gamma, mu, rsg);
  wout_prep_kernel<<<((NHID / 2) * 32 + 255) / 256, 256, 0, stream>>>(Wout, rsg,
                                                                      WpP);
  bout_prep_kernel<<<1, 32, 0, stream>>>(Wout, bout, beta, mu, rsg, bp);
  out_gemm_kernel<<<NROWS / 32, 128, 0, stream>>>(y1, WpP, bp, out);
}